// MaxMarginCriterion_77627238908605
// MI455X (gfx1250) — compile-verified
//
#include <hip/hip_runtime.h>
#include <hip/hip_bf16.h>
#include <stdint.h>

// MaxMarginCriterion on MI455X (gfx1250), compile-only tuning.
// N=16384 rows, C=4096 cols. Bandwidth-bound: ~537MB mandatory traffic.
// One workgroup per row; cossim row staged into LDS with CDNA5 async
// global->LDS copies overlapped with the one-hot target scan.

#define NROWS   16384
#define NCOLS   4096
#define MARGINF 0.1f
#define BLOCK   256
#define CHUNKS  4   // 4 x float4 per thread = 16 floats; 256 thr * 16 = 4096

__global__ void zero_out_kernel(float* out, int n) {
    int i = blockIdx.x * blockDim.x + threadIdx.x;
    if (i < n) out[i] = 0.0f;
}

__global__ __launch_bounds__(BLOCK)
void MaxMarginCriterion_kernel(const float* __restrict__ cossim,
                               const int*   __restrict__ target,
                               float*       __restrict__ out)
{
    __shared__ float sCos[NCOLS];   // 16 KB staged row
    __shared__ int   sIdx;
    __shared__ float sSum;

    const int tid = threadIdx.x;
    const int row = blockIdx.x;

    if (tid == 0) sSum = 0.0f;

    const float* rowC = cossim + (size_t)row * NCOLS;
    const int*   rowT = target + (size_t)row * NCOLS;

    // ---- Phase 1a: kick off async DMA of the cossim row into LDS.
    // Per instruction, each of 32 lanes moves 16B -> 512B/wave/instr.
    // Addresses are lane-contiguous per instruction: coalesced on the
    // global side, bank-conflict-free on the LDS side.
    // Low 32 bits of a generic shared pointer == LDS byte offset.
    uint32_t ldsBase = (uint32_t)(uintptr_t)(&sCos[0]);
#pragma unroll
    for (int i = 0; i < CHUNKS; ++i) {
        uint32_t off = (uint32_t)(i * (BLOCK * 16) + tid * 16);
        asm volatile("global_load_async_to_lds_b128 %0, %1, %2"
                     :
                     : "v"(ldsBase + off), "v"(off), "s"(rowC)
                     : "memory");
    }

    // ---- Phase 1b (overlapped): scan one-hot target row for correct col.
    const int4* tv = reinterpret_cast<const int4*>(rowT);
#pragma unroll
    for (int i = 0; i < CHUNKS; ++i) {
        int  idx4 = i * BLOCK + tid;          // coalesced int4 index
        int4 t    = tv[idx4];
        int  base = idx4 * 4;
        if (t.x == 1) sIdx = base + 0;
        if (t.y == 1) sIdx = base + 1;
        if (t.z == 1) sIdx = base + 2;
        if (t.w == 1) sIdx = base + 3;
    }

    // Drain this wave's async copies, then workgroup barrier so every
    // wave sees the fully-staged row and the published index.
    asm volatile("s_wait_asynccnt 0x0" ::: "memory");
    __syncthreads();

    // ---- Phase 2: hinge sum out of LDS.
    const float  sc = sCos[sIdx];             // LDS broadcast read
    const float4* sv = reinterpret_cast<const float4*>(sCos);
    float acc = 0.0f;
#pragma unroll
    for (int i = 0; i < CHUNKS; ++i) {
        float4 v = sv[i * BLOCK + tid];       // conflict-free ds_load_b128
        acc += fmaxf(MARGINF + v.x - sc, 0.0f);
        acc += fmaxf(MARGINF + v.y - sc, 0.0f);
        acc += fmaxf(MARGINF + v.z - sc, 0.0f);
        acc += fmaxf(MARGINF + v.w - sc, 0.0f);
    }
    // Correct column contributed exactly MARGINF; excluded by subtracting
    // it once per row at the end.

    // ---- Phase 3: wave32 reduction, then block, then global.
#pragma unroll
    for (int m = 16; m >= 1; m >>= 1)
        acc += __shfl_xor(acc, m, 32);

    if ((tid & 31) == 0) atomicAdd(&sSum, acc);
    __syncthreads();

    if (tid == 0)
        atomicAdd(out, (sSum - MARGINF) * (1.0f / (float)NROWS));
}

extern "C" void kernel_launch(void* const* d_in, const int* in_sizes, int n_in,
                              void* d_out, int out_size, void* d_ws, size_t ws_size,
                              hipStream_t stream) {
    (void)in_sizes; (void)n_in; (void)d_ws; (void)ws_size;
    const float* cossim = (const float*)d_in[0];
    const int*   target = (const int*)d_in[1];
    float*       out    = (float*)d_out;

    // d_out is poisoned; atomics accumulate, so zero it first (stream-ordered,
    // graph-capture safe).
    zero_out_kernel<<<1, 64, 0, stream>>>(out, out_size);
    MaxMarginCriterion_kernel<<<NROWS, BLOCK, 0, stream>>>(cossim, target, out);
}